// SAGPoolScore_35141422416138
// MI455X (gfx1250) — compile-verified
//
#include <hip/hip_runtime.h>

typedef float v2f __attribute__((ext_vector_type(2)));
typedef float v8f __attribute__((ext_vector_type(8)));

#define D          64
#define TILE       16
#define LDS_STRIDE 68   // 64 + 4 pad: bank stride 4 -> conflict-free column reads
#define WPB        8    // waves per 256-thread block (wave32)

// Kernel 1: per-node dual dot products via fp32 WMMA.
//   s_rel[n] = x[n] . W_rel        (scratch, consumed by edge kernel)
//   out[n]   = x[n] . W_root + b_rel   (init; edge kernel accumulates on top)
__global__ __launch_bounds__(256) void sag_dot_wmma(
    const float* __restrict__ x,
    const float* __restrict__ W_rel,
    const float* __restrict__ b_rel,
    const float* __restrict__ W_root,
    float* __restrict__ out,
    float* __restrict__ s_rel,
    int n_nodes)
{
    __shared__ float lds[WPB][TILE * LDS_STRIDE];

    const int lane = threadIdx.x & 31;
    const int wave = threadIdx.x >> 5;
    const int tile = (blockIdx.x * WPB + wave) * TILE;
    const bool active = (tile < n_nodes);

    if (active) {
        float* t = lds[wave];
        // Stage 16x64 fp32 tile, coalesced: 32 lanes x float2 per row.
#pragma unroll
        for (int r = 0; r < TILE; ++r) {
            int g = tile + r;
            if (g >= n_nodes) g = n_nodes - 1;           // clamp (N%16==0 in practice)
            v2f v = *(const v2f*)(x + (size_t)g * D + lane * 2);
            t[r * LDS_STRIDE + lane * 2 + 0] = v.x;
            t[r * LDS_STRIDE + lane * 2 + 1] = v.y;
        }
    }
    __syncthreads();          // uniform participation; per-wave LDS regions
    if (!active) return;      // wave-uniform exit -> EXEC all-1s for WMMA

    // Fragment coordinates (V_WMMA_F32_16X16X4_F32, wave32):
    //   A: lane L holds row M=L%16; VGPR0=K(2*(L/16)), VGPR1=K(2*(L/16)+1)
    //   B: lane L holds col N=L%16; same K striping as A
    const int   m     = lane & 15;
    const int   khalf = lane >> 4;
    const int   k0    = khalf * 2;
    const float bmask = (m < 2) ? 1.0f : 0.0f;          // only cols 0/1 carry weights
    const float* wsel = (m == 1) ? W_root : W_rel;
    const float* t    = lds[wave];

    v8f acc = {};
#pragma unroll
    for (int kb = 0; kb < D; kb += 4) {
        v2f a, b;
        a.x = t[m * LDS_STRIDE + kb + k0];
        a.y = t[m * LDS_STRIDE + kb + k0 + 1];
        b.x = wsel[kb + k0]     * bmask;                // tiny, L2/K$-resident
        b.y = wsel[kb + k0 + 1] * bmask;
        acc = __builtin_amdgcn_wmma_f32_16x16x4_f32(
            /*neg_a=*/false, a, /*neg_b=*/false, b,
            /*c_mod=*/(short)0, acc, /*reuse_a=*/false, /*reuse_b=*/false);
    }

    // D layout: lane L, VGPR v -> (M = v + 8*khalf, N = m). Col 0 = rel, col 1 = root.
    const float br    = b_rel[0];
    const int   mbase = khalf * 8;
    if (m == 0) {
#pragma unroll
        for (int v = 0; v < 8; ++v) {
            int g = tile + mbase + v;
            if (g < n_nodes) s_rel[g] = acc[v];
        }
    } else if (m == 1) {
#pragma unroll
        for (int v = 0; v < 8; ++v) {
            int g = tile + mbase + v;
            if (g < n_nodes) out[g] = acc[v] + br;
        }
    }
}

// Kernel 2: scalar-per-edge scatter (W_rel pushed through the linear segment_sum).
__global__ __launch_bounds__(256) void sag_edge_scatter(
    const long long* __restrict__ esrc,   // edge_index row 0 (int64)
    const long long* __restrict__ edst,   // edge_index row 1 (int64)
    const float* __restrict__ s_rel,
    float* __restrict__ out,
    int E)
{
    int i = blockIdx.x * blockDim.x + threadIdx.x;
    const int stride = gridDim.x * blockDim.x;
    for (; i < E; i += stride) {
        const int nxt = i + stride;
        if (nxt < E) {                       // global_prefetch_b8 for next chunk
            __builtin_prefetch(&esrc[nxt], 0, 1);
            __builtin_prefetch(&edst[nxt], 0, 1);
        }
        const int s = (int)__builtin_nontemporal_load(&esrc[i]);  // streamed indices
        const int d = (int)__builtin_nontemporal_load(&edst[i]);
        atomicAdd(&out[d], s_rel[s]);        // global_atomic_add_f32, L2-resident
    }
}

extern "C" void kernel_launch(void* const* d_in, const int* in_sizes, int n_in,
                              void* d_out, int out_size, void* d_ws, size_t ws_size,
                              hipStream_t stream) {
    const float*     x      = (const float*)d_in[0];
    const long long* ei     = (const long long*)d_in[1];   // [2, E] int64
    const float*     W_rel  = (const float*)d_in[2];
    const float*     b_rel  = (const float*)d_in[3];
    const float*     W_root = (const float*)d_in[4];
    float*           out    = (float*)d_out;

    const int n_nodes = in_sizes[0] / D;
    const int E       = in_sizes[1] / 2;
    float* s_rel = (float*)d_ws;                           // N floats = 400 KB scratch

    const int tiles   = (n_nodes + TILE - 1) / TILE;
    const int blocks1 = (tiles + WPB - 1) / WPB;
    sag_dot_wmma<<<blocks1, 256, 0, stream>>>(x, W_rel, b_rel, W_root, out, s_rel, n_nodes);

    // ~4 edges/thread so the grid-stride prefetch is live.
    const int blocks2 = (E + 1023) / 1024;
    sag_edge_scatter<<<blocks2, 256, 0, stream>>>(ei, ei + E, s_rel, out, E);
}